// SubSamplingLayer_62191126446417
// MI455X (gfx1250) — compile-verified
//
#include <hip/hip_runtime.h>
#include <math.h>

// Problem constants (match reference).
#define RES   128
#define PTS   4096
#define BATCH 8

typedef __attribute__((ext_vector_type(2))) float v2f;
typedef __attribute__((ext_vector_type(8))) float v8f;

// ---------------------------------------------------------------------------
// Kernel 1: build phasor tables  Ax[p][g] = exp(i*2pi/RES * kx_p * (g-64)),
//                                Ay[p][g] = exp(i*2pi/RES * ky_p * (g-64))
// Stored as float2 (cos, sin), row-major (p, g).  One thread per (p,g).
// ---------------------------------------------------------------------------
__global__ __launch_bounds__(256) void nufft_build_tables(
    const float* __restrict__ traj,
    float2* __restrict__ ax,
    float2* __restrict__ ay)
{
    int t = blockIdx.x * blockDim.x + threadIdx.x;   // 0 .. PTS*RES-1
    int p = t >> 7;
    int g = t & (RES - 1);
    float coord = (float)g - 64.0f;                  // centered grid coord
    const float ph = 6.28318530717958647692f / (float)RES;
    float kx = traj[2 * p + 0];
    float ky = traj[2 * p + 1];
    float sx, cx, sy, cy;
    __sincosf(ph * kx * coord, &sx, &cx);
    __sincosf(ph * ky * coord, &sy, &cy);
    ax[t] = make_float2(cx, sx);
    ay[t] = make_float2(cy, sy);
}

// ---------------------------------------------------------------------------
// Kernel 2: bilinear grid-sample (zero padding, align_corners=False) of the
// cartesian k-space at the trajectory points -> complex c[b][p].
// One thread per trajectory point, loops over the 8 batches.
// ---------------------------------------------------------------------------
__global__ __launch_bounds__(256) void nufft_sample_c(
    const float* __restrict__ ksp,     // (B,1,H,W,2) flat
    const float* __restrict__ traj,    // (P,2)
    float2* __restrict__ cout)         // (B,P)
{
    int p = blockIdx.x * blockDim.x + threadIdx.x;   // 0..PTS-1
    float kx = traj[2 * p + 0];
    float ky = traj[2 * p + 1];
    // pts_norm = traj/64 ; x = (norm+1)*64 - 0.5 = traj + 63.5
    float x = kx + 63.5f;
    float y = ky + 63.5f;
    float x0f = floorf(x);
    float y0f = floorf(y);
    float wx = x - x0f;
    float wy = y - y0f;
    int x0 = (int)x0f;
    int y0 = (int)y0f;

    int   xi[2] = { x0, x0 + 1 };
    int   yi[2] = { y0, y0 + 1 };
    float wxv[2] = { 1.0f - wx, wx };
    float wyv[2] = { 1.0f - wy, wy };

    for (int b = 0; b < BATCH; ++b) {
        const float2* img = (const float2*)ksp + (size_t)b * RES * RES;
        float re = 0.0f, im = 0.0f;
        #pragma unroll
        for (int ty = 0; ty < 2; ++ty) {
            #pragma unroll
            for (int tx = 0; tx < 2; ++tx) {
                bool v = (xi[tx] >= 0) & (xi[tx] < RES) & (yi[ty] >= 0) & (yi[ty] < RES);
                if (v) {
                    float  w = wxv[tx] * wyv[ty];
                    float2 s = img[yi[ty] * RES + xi[tx]];
                    re += w * s.x;
                    im += w * s.y;
                }
            }
        }
        cout[(size_t)b * PTS + p] = make_float2(re, im);
    }
}

// ---------------------------------------------------------------------------
// Kernel 3: complex GEMM via V_WMMA_F32_16X16X4_F32.
// img[b,h,w] = (1/RES) * sum_p (c[b,p]*Ay[p,h]) * Ax[p,w]
// One wave (32 lanes) per 16x16 output tile; K = 4096 in steps of 4.
// A-fragment layout (ISA 7.12.2, 32-bit 16x4): lane l, m=l&15, k0=2*(l>>4):
//   frag = (A[m][k0], A[m][k0+1]).   B 4x16: lane l, n=l&15, same k0 pair.
// C/D 16x16: vgpr j, lane l -> M = j + 8*(l>>4), N = l&15.
// ---------------------------------------------------------------------------
__global__ __launch_bounds__(32) void nufft_wmma_gemm(
    const float2* __restrict__ cArr,   // (B, P)
    const float2* __restrict__ ayT,    // (P, RES)  phasors for h axis
    const float2* __restrict__ axT,    // (P, RES)  phasors for w axis
    float2* __restrict__ out)          // (B, RES, RES) as (re,im)
{
    const int tile  = blockIdx.x;          // 0..63 : (hTile, wTile)
    const int b     = blockIdx.y;          // 0..7
    const int hBase = (tile >> 3) * 16;
    const int wBase = (tile & 7) * 16;
    const int l     = threadIdx.x;
    const int m     = l & 15;              // A row (h) / B col (w) / D col
    const int koff  = (l >> 4) * 2;        // which K-pair this half-wave holds

    v8f accRe = {0.f,0.f,0.f,0.f,0.f,0.f,0.f,0.f};
    v8f accIm = {0.f,0.f,0.f,0.f,0.f,0.f,0.f,0.f};

    const float2* cB  = cArr + (size_t)b * PTS;
    const float2* ayC = ayT + hBase + m;   // column h fixed per lane
    const float2* axC = axT + wBase + m;   // column w fixed per lane

    #pragma unroll 4
    for (int p0 = 0; p0 < PTS; p0 += 4) {
        const int pa = p0 + koff;
        // A-side: (c[b,p] * Ay[p,h]) for p = pa, pa+1
        float2 ay0 = ayC[(size_t)pa * RES];
        float2 ay1 = ayC[(size_t)(pa + 1) * RES];
        float2 c0  = cB[pa];
        float2 c1  = cB[pa + 1];
        float ar0 = c0.x * ay0.x - c0.y * ay0.y;
        float ai0 = c0.x * ay0.y + c0.y * ay0.x;
        float ar1 = c1.x * ay1.x - c1.y * ay1.y;
        float ai1 = c1.x * ay1.y + c1.y * ay1.x;
        // B-side: Ax[p,w] for p = pa, pa+1
        float2 ax0 = axC[(size_t)pa * RES];
        float2 ax1 = axC[(size_t)(pa + 1) * RES];

        v2f Ar  = { ar0,  ar1 };
        v2f Ai  = { ai0,  ai1 };
        v2f AiN = { -ai0, -ai1 };
        v2f Br  = { ax0.x, ax1.x };
        v2f Bi  = { ax0.y, ax1.y };

        // Re += Ar*Br - Ai*Bi ; Im += Ar*Bi + Ai*Br   (fp32 WMMA, K=4)
        accRe = __builtin_amdgcn_wmma_f32_16x16x4_f32(false, Ar,  false, Br,
                                                      (short)0, accRe, false, false);
        accRe = __builtin_amdgcn_wmma_f32_16x16x4_f32(false, AiN, false, Bi,
                                                      (short)0, accRe, false, false);
        accIm = __builtin_amdgcn_wmma_f32_16x16x4_f32(false, Ar,  false, Bi,
                                                      (short)0, accIm, false, false);
        accIm = __builtin_amdgcn_wmma_f32_16x16x4_f32(false, Ai,  false, Br,
                                                      (short)0, accIm, false, false);
    }

    const float s = 1.0f / (float)RES;
    const int w    = wBase + m;
    const int hTop = hBase + (l >> 4) * 8;
    #pragma unroll
    for (int j = 0; j < 8; ++j) {
        int h = hTop + j;
        out[((size_t)b * RES + h) * RES + w] = make_float2(accRe[j] * s, accIm[j] * s);
    }
}

// ---------------------------------------------------------------------------
extern "C" void kernel_launch(void* const* d_in, const int* in_sizes, int n_in,
                              void* d_out, int out_size, void* d_ws, size_t ws_size,
                              hipStream_t stream)
{
    (void)in_sizes; (void)n_in; (void)out_size; (void)ws_size;
    const float* ksp  = (const float*)d_in[0];   // (8,1,128,128,2) f32
    const float* traj = (const float*)d_in[1];   // (4096,2) f32

    // Workspace layout: Ax table | Ay table | c
    char*   ws = (char*)d_ws;
    float2* ax = (float2*)ws;                                           // 4 MB
    float2* ay = (float2*)(ws + (size_t)PTS * RES * sizeof(float2));    // 4 MB
    float2* c  = (float2*)(ws + 2 * (size_t)PTS * RES * sizeof(float2));// 256 KB

    nufft_build_tables<<<(PTS * RES) / 256, 256, 0, stream>>>(traj, ax, ay);
    nufft_sample_c<<<PTS / 256, 256, 0, stream>>>(ksp, traj, c);
    nufft_wmma_gemm<<<dim3(64, BATCH), 32, 0, stream>>>(c, ay, ax, (float2*)d_out);
}